// VQLayer_57286273794526
// MI455X (gfx1250) — compile-verified
//
#include <hip/hip_runtime.h>
#include <hip/hip_bf16.h>

#define DIM      256
#define NCODES   16384
#define NROWS    16384
#define KCHUNKS  (DIM / 4)      // 64 wmma_f32_16x16x4 steps per 16x16 tile
#define BSTRIDE  260            // padded LDS stride (floats) -> conflict-free ds_load_b64

typedef __attribute__((ext_vector_type(2))) float v2f;
typedef __attribute__((ext_vector_type(8))) float v8f;

// ---------------------------------------------------------------------------
// Kernel 0: ||e_k||^2 per code (one wave per code) + zero usage histogram
// ---------------------------------------------------------------------------
__global__ __launch_bounds__(256) void vq_enorm_kernel(
    const float* __restrict__ embed, float* __restrict__ enorm,
    int* __restrict__ counts) {
  const int lane = threadIdx.x & 31;
  const int wave = threadIdx.x >> 5;
  const int code = blockIdx.x * 8 + wave;            // 2048 blocks * 8 waves = 16384
  const float* e = embed + (size_t)code * DIM;
  float s = 0.f;
#pragma unroll
  for (int i = 0; i < 8; ++i) {
    float v = e[i * 32 + lane];
    s += v * v;
  }
#pragma unroll
  for (int off = 16; off >= 1; off >>= 1) s += __shfl_xor(s, off, 32);
  if (lane == 0) enorm[code] = s;

  int gid = blockIdx.x * 256 + threadIdx.x;
  if (gid < NCODES) counts[gid] = 0;
}

// ---------------------------------------------------------------------------
// Kernel 1: fused GEMM (latent . embed^T) + argmin, on v_wmma_f32_16x16x4_f32.
// Block = 8 waves; wave owns 16 rows (A in registers); all waves share a
// 16-code embed tile staged in LDS each iteration.
// ---------------------------------------------------------------------------
__global__ __launch_bounds__(256) void vq_argmin_kernel(
    const float* __restrict__ x, const float* __restrict__ embed,
    const float* __restrict__ enorm, int* __restrict__ idx_out) {
  __shared__ float Bt[16 * BSTRIDE];                 // 16.25 KB

  const int lane   = threadIdx.x & 31;
  const int wave   = threadIdx.x >> 5;
  const int m_base = blockIdx.x * 128 + wave * 16;   // 128 blocks cover 16384 rows
  const int mrow   = lane & 15;
  const int koff   = (lane < 16) ? 0 : 2;            // wmma f32 A/B lane layout

  // Preload this wave's A fragments (16 rows x 256 d) into registers.
  const float* xr = x + (size_t)(m_base + mrow) * DIM;
  v2f a[KCHUNKS];
#pragma unroll
  for (int kc = 0; kc < KCHUNKS; ++kc) {
    a[kc].x = xr[4 * kc + koff];
    a[kc].y = xr[4 * kc + koff + 1];
  }

  float rmin[8];
  int   ridx[8];
#pragma unroll
  for (int v = 0; v < 8; ++v) { rmin[v] = 3.4e38f; ridx[v] = 0; }

  for (int tile = 0; tile < NCODES / 16; ++tile) {
    __syncthreads();  // previous iteration's reads of Bt are done
    {
      // Cooperative stage: 16 codes x 256 floats, thread t -> code t/16, 16 floats
      const int code = threadIdx.x >> 4;
      const int ks   = (threadIdx.x & 15) * 16;
      const float4* src = (const float4*)(embed + ((size_t)(tile * 16 + code)) * DIM + ks);
      float4* dst = (float4*)(&Bt[code * BSTRIDE + ks]);
      dst[0] = src[0]; dst[1] = src[1]; dst[2] = src[2]; dst[3] = src[3];
    }
    __syncthreads();

    v8f c = {0.f, 0.f, 0.f, 0.f, 0.f, 0.f, 0.f, 0.f};
#pragma unroll
    for (int kc = 0; kc < KCHUNKS; ++kc) {
      const float* bp = &Bt[mrow * BSTRIDE + 4 * kc + koff];
      v2f b;
      b.x = bp[0];
      b.y = bp[1];
      c = __builtin_amdgcn_wmma_f32_16x16x4_f32(
          /*neg_a=*/false, a[kc], /*neg_b=*/false, b,
          /*c_mod=*/(short)0, c, /*reuse_a=*/false, /*reuse_b=*/false);
    }

    const int   code = tile * 16 + mrow;             // this lane's column (code)
    const float en   = enorm[code];
#pragma unroll
    for (int v = 0; v < 8; ++v) {
      float d = en - 2.0f * c[v];                    // ||z||^2 constant per row: dropped
      if (d < rmin[v]) { rmin[v] = d; ridx[v] = code; }
    }
  }

  // Cross-lane argmin within each 16-lane half (strides <= 8 keep halves apart).
#pragma unroll
  for (int v = 0; v < 8; ++v) {
    float val = rmin[v];
    int   id  = ridx[v];
#pragma unroll
    for (int off = 8; off >= 1; off >>= 1) {
      float oval = __shfl_xor(val, off, 32);
      int   oid  = __shfl_xor(id, off, 32);
      if (oval < val || (oval == val && oid < id)) { val = oval; id = oid; }
    }
    if (lane == 0)  idx_out[m_base + v]     = id;    // rows M = v      (lanes 0-15)
    if (lane == 16) idx_out[m_base + 8 + v] = id;    // rows M = v + 8  (lanes 16-31)
  }
}

// ---------------------------------------------------------------------------
// Kernel 2: gather x_q = embed[ind], loss partials (deterministic), histogram
// ---------------------------------------------------------------------------
__global__ __launch_bounds__(256) void vq_gather_kernel(
    const float* __restrict__ x, const float* __restrict__ embed,
    const int* __restrict__ idx, float* __restrict__ xq,
    float* __restrict__ indf, int* __restrict__ counts,
    float* __restrict__ partial) {
  __shared__ float red[256];
  const int row = blockIdx.x * 16 + (threadIdx.x >> 4);  // 1024 blocks
  const int seg = threadIdx.x & 15;
  const int id  = idx[row];

  const float4* e  = (const float4*)(embed + (size_t)id * DIM + seg * 16);
  const float4* xr = (const float4*)(x + (size_t)row * DIM + seg * 16);
  float4*       q  = (float4*)(xq + (size_t)row * DIM + seg * 16);

  float s = 0.f;
#pragma unroll
  for (int i = 0; i < 4; ++i) {
    float4 ev = e[i], xv = xr[i];
    float dx = ev.x - xv.x, dy = ev.y - xv.y, dz = ev.z - xv.z, dw = ev.w - xv.w;
    s += dx * dx + dy * dy + dz * dz + dw * dw;
    q[i] = ev;
  }
  if (seg == 0) {
    atomicAdd(&counts[id], 1);                       // integer atomic: deterministic
    indf[row] = (float)id;
  }
  red[threadIdx.x] = s;
  __syncthreads();
  for (int off = 128; off >= 1; off >>= 1) {
    if (threadIdx.x < off) red[threadIdx.x] += red[threadIdx.x + off];
    __syncthreads();
  }
  if (threadIdx.x == 0) partial[blockIdx.x] = red[0];
}

// ---------------------------------------------------------------------------
// Kernel 3: fixed-order reduction -> quant_loss, unused_codes
// ---------------------------------------------------------------------------
__global__ __launch_bounds__(256) void vq_finalize_kernel(
    const float* __restrict__ partial, const int* __restrict__ counts,
    float* __restrict__ out_scalars) {
  __shared__ float fs[256];
  __shared__ int   us[256];
  const int t = threadIdx.x;
  float s = 0.f;
  for (int i = t; i < 1024; i += 256) s += partial[i];
  int u = 0;
  for (int i = t; i < NCODES; i += 256) u += (counts[i] == 0) ? 1 : 0;
  fs[t] = s; us[t] = u;
  __syncthreads();
  for (int off = 128; off >= 1; off >>= 1) {
    if (t < off) { fs[t] += fs[t + off]; us[t] += us[t + off]; }
    __syncthreads();
  }
  if (t == 0) {
    // loss = mean(dq^2) + 0.25*mean(dq^2) = 1.25 * sum / (N*D)
    out_scalars[0] = 1.25f * fs[0] / (float)(NROWS * DIM);
    out_scalars[1] = (float)us[0];
  }
}

// ---------------------------------------------------------------------------
extern "C" void kernel_launch(void* const* d_in, const int* in_sizes, int n_in,
                              void* d_out, int out_size, void* d_ws, size_t ws_size,
                              hipStream_t stream) {
  const float* x     = (const float*)d_in[0];   // (128,128,256) f32
  const float* embed = (const float*)d_in[1];   // (16384,256)   f32
  float* out = (float*)d_out;

  char*  ws      = (char*)d_ws;
  float* enorm   = (float*)(ws);                   // 16384 f
  int*   idx     = (int*)  (ws + 65536);           // 16384 i32
  int*   counts  = (int*)  (ws + 131072);          // 16384 i32
  float* partial = (float*)(ws + 196608);          // 1024 f

  float* xq      = out;                            // 4194304 floats
  float* scalars = out + (size_t)NROWS * DIM;      // [loss, unused]
  float* indf    = scalars + 2;                    // 16384 floats

  vq_enorm_kernel<<<2048, 256, 0, stream>>>(embed, enorm, counts);
  vq_argmin_kernel<<<128, 256, 0, stream>>>(x, embed, enorm, idx);
  vq_gather_kernel<<<1024, 256, 0, stream>>>(x, embed, idx, xq, indf, counts, partial);
  vq_finalize_kernel<<<1, 256, 0, stream>>>(partial, counts, scalars);
}